// TokenSelection_6708738917103
// MI455X (gfx1250) — compile-verified
//
#include <hip/hip_runtime.h>
#include <math.h>

// Problem constants (reference: B=4, S=8192, D=1024, NSEL=16)
#define NB    4
#define SDIM  8192
#define DDIM  1024
#define NSEL  16
#define NTOK  (NB * SDIM)   // 32768 tokens

typedef __attribute__((ext_vector_type(2))) float v2f;
typedef __attribute__((ext_vector_type(8))) float v8f;

// Sum across the 16-lane group (xor bits 0..3 stay inside the group).
__device__ __forceinline__ float grp16_sum(float v) {
    v += __shfl_xor(v, 1, 32);
    v += __shfl_xor(v, 2, 32);
    v += __shfl_xor(v, 4, 32);
    v += __shfl_xor(v, 8, 32);
    return v;
}

// ---------------------------------------------------------------------------
// Kernel 1: scores GEMM (WMMA f32 16x16x4) + bias + PReLU + celu*silu+gelu,
// then reduce over n with selu(token_weights) -> combined[t], and plain sum
// over n -> src[t].  One wave per 16-token tile, 8 waves per block.
//
// K-permutation trick: WMMA#1 window = {k0,k0+1, k0+4,k0+5}, WMMA#2 window =
// {k0+2,k0+3, k0+6,k0+7}.  Lane half h loads its own float4 at column
// k0 + 4h and feeds .xy to WMMA#1, .zw to WMMA#2 -- no cross-lane exchange.
// B is stored in LDS pre-permuted to match:
//   k = 8c + 4h + 2j + e  ->  sW[c*128 + j*64 + h*32 + 2n + e]
// so each WMMA's B read hits banks {h*32+2n, h*32+2n+1}: conflict-free, and
// the j=0 / j=1 reads (256 B apart) fuse into one ds_load_2addr_b64.
// ---------------------------------------------------------------------------
__global__ __launch_bounds__(256) void ts_gemm_act_kernel(
    const float* __restrict__ x, const float* __restrict__ W,
    const float* __restrict__ bvec, const float* __restrict__ aptr,
    const float* __restrict__ tw,
    float* __restrict__ combined, float* __restrict__ srcOut)
{
    __shared__ float sW[NSEL * DDIM];   // 64 KB
    const int tid = threadIdx.x;
    for (int idx = tid; idx < NSEL * DDIM; idx += 256) {
        const int n = idx >> 10;            // selection
        const int k = idx & (DDIM - 1);     // feature
        const int c = k >> 3;
        const int h = (k >> 2) & 1;
        const int j = (k >> 1) & 1;
        const int e = k & 1;
        sW[c * 128 + j * 64 + h * 32 + n * 2 + e] = W[idx];
    }
    __syncthreads();

    const int lane = tid & 31;
    const int wave = tid >> 5;
    const int l    = lane & 15;             // n-column / A-row within half
    const int half = lane >> 4;             // K-window half
    const int tileBase = (blockIdx.x * 8 + wave) * 16;

    const float*  xq  = x + (size_t)(tileBase + l) * DDIM + half * 4;
    const float2* sW2 = (const float2*)sW;
    const int     bB  = half * 16 + l;      // float2 slot within a chunk

    v8f acc = {};
    #pragma unroll 4
    for (int k0 = 0; k0 < DDIM; k0 += 8) {
        const int c = k0 >> 3;
        const float4 q   = *(const float4*)(xq + k0);   // own quad only
        const float2 b0f = sW2[c * 64 + bB];            // j = 0
        const float2 b1f = sW2[c * 64 + 32 + bB];       // j = 1

        v2f a0; a0.x = q.x;   a0.y = q.y;
        v2f a1; a1.x = q.z;   a1.y = q.w;
        v2f b0; b0.x = b0f.x; b0.y = b0f.y;
        v2f b1; b1.x = b1f.x; b1.y = b1f.y;

        acc = __builtin_amdgcn_wmma_f32_16x16x4_f32(false, a0, false, b0,
                                                    (short)0, acc, false, false);
        acc = __builtin_amdgcn_wmma_f32_16x16x4_f32(false, a1, false, b1,
                                                    (short)0, acc, false, false);
    }

    // Epilogue: lane holds D[M = r + 8*half][N = l] in acc[r].
    const float bias   = bvec[l];
    const float aSlope = aptr[0];
    const float twv    = tw[l];
    const float SELU_SCALE = 1.0507009873554805f;
    const float SELU_ALPHA = 1.6732632423543772f;
    const float seluTw = SELU_SCALE *
        (twv > 0.f ? twv : SELU_ALPHA * (expf(twv) - 1.f));

    float comb[8], ssum[8];
    #pragma unroll
    for (int r = 0; r < 8; ++r) {
        const float s  = acc[r] + bias;
        const float p  = s >= 0.f ? s : aSlope * s;                 // PReLU
        const float ce = p > 0.f ? p : (expf(p) - 1.f);             // celu(a=1)
        const float si = p / (1.f + expf(-p));                      // silu
        const float ge = 0.5f * p * (1.f + erff(p * 0.7071067811865476f)); // gelu
        const float ts = ce * si + ge;
        comb[r] = grp16_sum(ts * seluTw);   // sum over n (lanes of the group)
        ssum[r] = grp16_sum(ts);
    }
    // After the butterfly every lane of a 16-group holds the row sums; let
    // lane l==r of each half write row m = r + 8*half.
    #pragma unroll
    for (int r = 0; r < 8; ++r) {
        if (l == r) {
            const int m = half * 8 + r;
            combined[tileBase + m] = comb[r];
            srcOut[tileBase + m]   = ssum[r];
        }
    }
}

// ---------------------------------------------------------------------------
// Kernel 2: rank(j) = #{i : combined[i] > combined[j], stable ties}, then
// importance[j] = 1 + src[rank(j)]   (argsort(-combined) scatter-add of ones
// collapses to exactly this because idx is a permutation).
// ---------------------------------------------------------------------------
__global__ __launch_bounds__(256) void ts_rank_kernel(
    const float* __restrict__ combined, const float* __restrict__ src,
    float* __restrict__ importance)
{
    const int t    = blockIdx.x * 256 + threadIdx.x;  // global token
    const int brow = t >> 13;                         // /SDIM
    const int j    = t & (SDIM - 1);
    const float* row = combined + brow * SDIM;
    const float  v   = row[j];

    const float4* row4 = (const float4*)row;
    int rank = 0;
    for (int i4 = 0; i4 < SDIM / 4; ++i4) {
        const float4 c = row4[i4];
        const int i = i4 * 4;
        rank += (c.x > v) || (c.x == v && (i + 0) < j);
        rank += (c.y > v) || (c.y == v && (i + 1) < j);
        rank += (c.z > v) || (c.z == v && (i + 2) < j);
        rank += (c.w > v) || (c.w == v && (i + 3) < j);
    }
    importance[t] = 1.0f + src[brow * SDIM + rank];
}

// ---------------------------------------------------------------------------
// Kernel 3: out = 1 + softmax(importance) per row. One block per batch row.
// ---------------------------------------------------------------------------
__global__ __launch_bounds__(1024) void ts_softmax_kernel(
    const float* __restrict__ imp, float* __restrict__ out)
{
    __shared__ float red[32];
    const int row = blockIdx.x;
    const int tid = threadIdx.x;
    const float* rp = imp + row * SDIM;

    float vals[8];
    float m = -3.402823466e38f;
    #pragma unroll
    for (int t = 0; t < 8; ++t) {
        vals[t] = rp[tid + t * 1024];
        m = fmaxf(m, vals[t]);
    }
    for (int w = 1; w < 32; w <<= 1) m = fmaxf(m, __shfl_xor(m, w, 32));
    if ((tid & 31) == 0) red[tid >> 5] = m;
    __syncthreads();
    if (tid < 32) {
        float v = red[tid];
        for (int w = 1; w < 32; w <<= 1) v = fmaxf(v, __shfl_xor(v, w, 32));
        red[tid] = v;
    }
    __syncthreads();
    m = red[0];

    float s = 0.f;
    #pragma unroll
    for (int t = 0; t < 8; ++t) { vals[t] = expf(vals[t] - m); s += vals[t]; }
    for (int w = 1; w < 32; w <<= 1) s += __shfl_xor(s, w, 32);
    __syncthreads();                       // red reuse
    if ((tid & 31) == 0) red[tid >> 5] = s;
    __syncthreads();
    if (tid == 0) {
        float v = 0.f;
        for (int i = 0; i < 32; ++i) v += red[i];
        red[0] = v;
    }
    __syncthreads();
    const float inv = 1.f / red[0];
    #pragma unroll
    for (int t = 0; t < 8; ++t)
        out[row * SDIM + tid + t * 1024] = 1.0f + vals[t] * inv;
}

// ---------------------------------------------------------------------------
extern "C" void kernel_launch(void* const* d_in, const int* in_sizes, int n_in,
                              void* d_out, int out_size, void* d_ws, size_t ws_size,
                              hipStream_t stream)
{
    const float* x    = (const float*)d_in[0];  // (B,S,D)
    const float* W    = (const float*)d_in[1];  // (NSEL,D)
    const float* bvec = (const float*)d_in[2];  // (NSEL,)
    const float* a    = (const float*)d_in[3];  // scalar
    const float* tw   = (const float*)d_in[4];  // (NSEL,)
    float* out        = (float*)d_out;          // (B,S,1) fp32

    float* combined   = (float*)d_ws;           // NTOK floats
    float* src        = combined + NTOK;        // NTOK floats
    float* importance = src + NTOK;             // NTOK floats

    ts_gemm_act_kernel<<<NTOK / (16 * 8), 256, 0, stream>>>(
        x, W, bvec, a, tw, combined, src);
    ts_rank_kernel<<<NTOK / 256, 256, 0, stream>>>(combined, src, importance);
    ts_softmax_kernel<<<NB, 1024, 0, stream>>>(importance, out);
}